// PatchEmbedding_49400713838690
// MI455X (gfx1250) — compile-verified
//
#include <hip/hip_runtime.h>
#include <hip/hip_bf16.h>

// v2f = A/B fragment of V_WMMA_F32_16X16X4_F32 (16x4 f32 over 32 lanes -> 2 VGPRs)
typedef float v2f __attribute__((ext_vector_type(2)));
typedef float v8f __attribute__((ext_vector_type(8)));

// ---------------------------------------------------------------------------
// Kernel 1: 2-level Haar DWT of x (B,3,224,224) fused with coeffs_to_array +
// im2col pack. Each thread owns one 4x4 input block (i,j in [0,56)) and emits
// the 16 coefficients that depend on it, scattered straight into the GEMM A
// matrix: A[M][K], M = b*196 + ph*14 + pw, K = c*256 + u*16 + v.
// ---------------------------------------------------------------------------
__global__ void haar2_pack_kernel(const float* __restrict__ x,
                                  float* __restrict__ A, int B) {
    int tid = blockIdx.x * blockDim.x + threadIdx.x;
    int total = B * 3 * 56 * 56;
    if (tid >= total) return;
    int j = tid % 56;
    int i = (tid / 56) % 56;
    int c = (tid / (56 * 56)) % 3;
    int b = tid / (56 * 56 * 3);

    const float* xp = x + ((size_t)(b * 3 + c) * 224 + (size_t)4 * i) * 224 + 4 * j;
    float r[4][4];
#pragma unroll
    for (int di = 0; di < 4; ++di) {
        float4 v = *(const float4*)(xp + (size_t)di * 224);
        r[di][0] = v.x; r[di][1] = v.y; r[di][2] = v.z; r[di][3] = v.w;
    }

    size_t Mbase = (size_t)b * 196;
    auto put = [&](int R, int Cc, float val) {
        int ph = R >> 4, pw = Cc >> 4, u = R & 15, v = Cc & 15;
        size_t M = Mbase + (size_t)(ph * 14 + pw);
        int K = c * 256 + u * 16 + v;
        A[M * 768 + K] = val;
    };

    float A1[2][2];
#pragma unroll
    for (int di = 0; di < 2; ++di)
#pragma unroll
        for (int dj = 0; dj < 2; ++dj) {
            float aa = r[2 * di][2 * dj];
            float bb = r[2 * di][2 * dj + 1];
            float cc = r[2 * di + 1][2 * dj];
            float dd = r[2 * di + 1][2 * dj + 1];
            A1[di][dj] = (aa + bb + cc + dd) * 0.5f;            // cA1
            put(2 * i + di, 112 + 2 * j + dj, (aa + bb - cc - dd) * 0.5f); // cH1
            put(112 + 2 * i + di, 2 * j + dj, (aa - bb + cc - dd) * 0.5f); // cV1
            put(112 + 2 * i + di, 112 + 2 * j + dj, (aa - bb - cc + dd) * 0.5f); // cD1
        }
    float aa = A1[0][0], bb = A1[0][1], cc = A1[1][0], dd = A1[1][1];
    put(i, j,           (aa + bb + cc + dd) * 0.5f);  // cA2
    put(i, 56 + j,      (aa + bb - cc - dd) * 0.5f);  // cH2
    put(56 + i, j,      (aa - bb + cc - dd) * 0.5f);  // cV2
    put(56 + i, 56 + j, (aa - bb - cc + dd) * 0.5f);  // cD2
}

// ---------------------------------------------------------------------------
// Kernel 2: fp32 WMMA GEMM. out[M][N] = A[M][K] * W[N][K]^T + bias[N]
// M = 25088, N = 768, K = 768. Block tile 128x128, 8 waves (2x4), wave tile
// 64x32 (4x2 accumulators of v8f). LDS stride padded to 20 floats: 16B-aligned
// float4 stores, conflict-free ds_load_b64 fragment reads.
// ---------------------------------------------------------------------------
#define LDS_STRIDE 20

__global__ __launch_bounds__(256) void wmma_gemm_kernel(
    const float* __restrict__ A, const float* __restrict__ Wt,
    const float* __restrict__ bias, float* __restrict__ out) {
    __shared__ float As[128 * LDS_STRIDE];
    __shared__ float Bs[128 * LDS_STRIDE];

    int m0 = blockIdx.x * 128;
    int n0 = blockIdx.y * 128;
    int t = threadIdx.x;
    int lane = t & 31;
    int w = t >> 5;
    int l = lane & 15;       // N (or M for A-frag) position within tile
    int h = lane >> 4;       // lane-group: selects k pair / C-row block
    int wm = (w & 1) * 64;   // wave M offset in block
    int wn = (w >> 1) * 32;  // wave N offset in block

    v8f acc[4][2];
#pragma unroll
    for (int tt = 0; tt < 4; ++tt)
#pragma unroll
        for (int ss = 0; ss < 2; ++ss) acc[tt][ss] = (v8f)0.0f;

    for (int k0 = 0; k0 < 768; k0 += 16) {
        // --- cooperative staging: 128 rows x 16 k of A and B^T (W rows) ----
#pragma unroll
        for (int rep = 0; rep < 2; ++rep) {
            int f = t + rep * 256;       // 0..511 float4 slots
            int row = f >> 2;            // 0..127
            int kq = (f & 3) * 4;        // 0,4,8,12
            float4 av = *(const float4*)(A + (size_t)(m0 + row) * 768 + k0 + kq);
            *(float4*)(&As[row * LDS_STRIDE + kq]) = av;
            float4 bv = *(const float4*)(Wt + (size_t)(n0 + row) * 768 + k0 + kq);
            *(float4*)(&Bs[row * LDS_STRIDE + kq]) = bv;
        }
        // prefetch next K slice into cache hierarchy (global_prefetch_b8)
        if (k0 + 16 < 768) {
            __builtin_prefetch(A + (size_t)(m0 + (t >> 1)) * 768 + k0 + 16, 0, 0);
            __builtin_prefetch(Wt + (size_t)(n0 + (t >> 1)) * 768 + k0 + 16, 0, 0);
        }
        __syncthreads();

        // --- 4 k-substeps of 4, 8 WMMAs each ------------------------------
#pragma unroll
        for (int kk = 0; kk < 4; ++kk) {
            int koff = kk * 4 + 2 * h;   // fragment element k = 2*h + vgpr
            v2f af[4], bf[2];
#pragma unroll
            for (int tt = 0; tt < 4; ++tt)
                af[tt] = *(const v2f*)(&As[(wm + tt * 16 + l) * LDS_STRIDE + koff]);
#pragma unroll
            for (int ss = 0; ss < 2; ++ss)
                bf[ss] = *(const v2f*)(&Bs[(wn + ss * 16 + l) * LDS_STRIDE + koff]);
#pragma unroll
            for (int tt = 0; tt < 4; ++tt)
#pragma unroll
                for (int ss = 0; ss < 2; ++ss)
                    acc[tt][ss] = __builtin_amdgcn_wmma_f32_16x16x4_f32(
                        false, af[tt], false, bf[ss], (short)0, acc[tt][ss],
                        false, false);
        }
        __syncthreads();
    }

    // --- epilogue: bias add + store. D layout: row = r + 8*h, col = l ------
#pragma unroll
    for (int ss = 0; ss < 2; ++ss) {
        int n = n0 + wn + ss * 16 + l;
        float bv = bias[n];
#pragma unroll
        for (int tt = 0; tt < 4; ++tt) {
#pragma unroll
            for (int r = 0; r < 8; ++r) {
                int m = m0 + wm + tt * 16 + 8 * h + r;
                out[(size_t)m * 768 + n] = acc[tt][ss][r] + bv;
            }
        }
    }
}

// ---------------------------------------------------------------------------
extern "C" void kernel_launch(void* const* d_in, const int* in_sizes, int n_in,
                              void* d_out, int out_size, void* d_ws, size_t ws_size,
                              hipStream_t stream) {
    const float* x    = (const float*)d_in[0];  // (B,3,224,224)
    const float* Wt   = (const float*)d_in[1];  // (768,3,16,16) == (768,768) row-major
    const float* bias = (const float*)d_in[2];  // (768,)
    float* out = (float*)d_out;                 // (B,196,768) == (M,768)
    float* Apk = (float*)d_ws;                  // packed A: M x 768 fp32 (~77 MB)

    int B = in_sizes[0] / (3 * 224 * 224);      // 128
    int total = B * 3 * 56 * 56;
    haar2_pack_kernel<<<(total + 255) / 256, 256, 0, stream>>>(x, Apk, B);

    int M = B * 196;                            // 25088, divisible by 128
    dim3 grid(M / 128, 768 / 128);
    wmma_gemm_kernel<<<grid, 256, 0, stream>>>(Apk, Wt, bias, out);
}